// AttentionBlock_90795608637622
// MI455X (gfx1250) — compile-verified
//
#include <hip/hip_runtime.h>
#include <hip/hip_bf16.h>
#include <math.h>

// ---------------------------------------------------------------------------
// Attention block with COLUMN-wise softmax (reference softmaxes over the
// query axis j), head dim 16, fp32 throughout.
//   out[:, :, 0:512]   = inputs            (fused into the projection pass)
//   out[:, :, 512:528] = sum_{i<=j} exp(s[j,i]/4 - M[i]) / Z[i] * v[i,:]
// Matrix work uses V_WMMA_F32_16X16X4_F32 (CDNA5 fp32 WMMA path).
//
// Memory-bound workload (~6 GFLOP vs ~132 MB compulsory traffic): the input
// is read exactly once (projection + passthrough-copy fused), q/k/v (6 MB)
// live in the 192 MB L2 for the two attention passes, and the 256 MB score
// matrix is never materialized (recomputed, ~2 GFLOP extra — free).
// ---------------------------------------------------------------------------

typedef float v2f __attribute__((ext_vector_type(2)));
typedef float v8f __attribute__((ext_vector_type(8)));

#define WMMA_F32X4(A, Bm, Cm) \
    __builtin_amdgcn_wmma_f32_16x16x4_f32(false, (A), false, (Bm), (short)0, (Cm), false, false)

constexpr int B_   = 16;
constexpr int T_   = 2048;
constexpr int C_   = 512;
constexpr int HD_  = 16;           // KS == VS == 16
constexpr int OUTC_ = C_ + HD_;    // 528
constexpr int NT_  = T_ / 16;      // 128 tiles along sequence
constexpr float INV_SQRT_KS = 0.25f;
constexpr float NEG_BIG = -1e30f;  // mask sentinel: exp underflows to 0

static __device__ __forceinline__ v2f ld2(const float* p) { return *(const v2f*)p; }

// ---------------------------------------------------------------------------
// Kernel 1: q/k/v = x @ W^T + b via WMMA f32 16x16x4, one wave per 16 rows,
// with the inputs->out[:, :, 0:512] passthrough copy fused in (the wave reads
// its whole 16x512 input tile exactly once across the k-loop, so each loaded
// A-chunk is also stored straight to the output — saves a 64 MB read pass).
// A(16x4):  lane -> M = lane&15, K = 2*(lane>>4) + vgpr
// B(4x16):  lane -> N = lane&15, K = 2*(lane>>4) + vgpr  (B[k][n] = W[n][k])
// D(16x16): lane -> N = lane&15, M = vgpr + 8*(lane>>4)
// ---------------------------------------------------------------------------
__global__ void qkv_proj_copy_kernel(const float* __restrict__ x,
                                     const float* __restrict__ Wq, const float* __restrict__ bq,
                                     const float* __restrict__ Wk, const float* __restrict__ bk,
                                     const float* __restrict__ Wv, const float* __restrict__ bv,
                                     float* __restrict__ qb, float* __restrict__ kb,
                                     float* __restrict__ vb, float* __restrict__ out) {
    const int lane = threadIdx.x;
    const int m0   = blockIdx.x * 16;          // global row block in [0, B*T)
    const int n    = lane & 15;
    const int koff = (lane >> 4) * 2;
    const int rhi  = 8 * (lane >> 4);

    const float* xrow = x   + (size_t)(m0 + n) * C_ + koff;
    float*       orow = out + (size_t)(m0 + n) * OUTC_ + koff;   // 8B aligned
    const float* wqr  = Wq + (size_t)n * C_ + koff;
    const float* wkr  = Wk + (size_t)n * C_ + koff;
    const float* wvr  = Wv + (size_t)n * C_ + koff;

    v8f qa = {}; v8f ka = {}; v8f va = {};
    for (int kc = 0; kc < C_; kc += 4) {
        v2f a   = ld2(xrow + kc);
        v2f bqv = ld2(wqr + kc);
        v2f bkv = ld2(wkr + kc);
        v2f bvv = ld2(wvr + kc);
        *(v2f*)(orow + kc) = a;                // fused passthrough copy
        qa = WMMA_F32X4(a, bqv, qa);
        ka = WMMA_F32X4(a, bkv, ka);
        va = WMMA_F32X4(a, bvv, va);
    }
    const float bqs = bq[n], bks = bk[n], bvs = bv[n];
#pragma unroll
    for (int r = 0; r < 8; ++r) {
        int row = m0 + r + rhi;
        qb[row * HD_ + n] = qa[r] + bqs;
        kb[row * HD_ + n] = ka[r] + bks;
        vb[row * HD_ + n] = va[r] + bvs;
    }
}

// ---------------------------------------------------------------------------
// Kernel 2: per-column (i) online max/sum over valid rows j >= i.
// One wave per (batch, i-tile). D-layout puts column i in lane (lane&15),
// rows split across vgprs + the two lane halves -> shfl_xor(16) merge.
// ---------------------------------------------------------------------------
__global__ void colsoftmax_stats_kernel(const float* __restrict__ qb,
                                        const float* __restrict__ kb,
                                        float* __restrict__ Mb, float* __restrict__ Zr) {
    const int lane  = threadIdx.x;
    const int b     = blockIdx.x / NT_;
    const int itile = blockIdx.x % NT_;
    const int i0    = itile * 16;
    const int base  = b * T_;
    const int n     = lane & 15;
    const int koff  = (lane >> 4) * 2;
    const int rhi   = 8 * (lane >> 4);
    const int iabs  = i0 + n;

    // B-matrix (K^T) chunks: B[k=d][n=i] = kb[(base+i0+n)*16 + d]
    const float* krow = kb + (size_t)(base + i0 + n) * HD_ + koff;
    v2f kB0 = ld2(krow + 0), kB1 = ld2(krow + 4), kB2 = ld2(krow + 8), kB3 = ld2(krow + 12);

    float m = NEG_BIG;
    float z = 0.0f;

    for (int jt = itile; jt < NT_; ++jt) {
        const int j0 = jt * 16;
        const float* qrow = qb + (size_t)(base + j0 + n) * HD_ + koff;
        v2f qA0 = ld2(qrow + 0), qA1 = ld2(qrow + 4), qA2 = ld2(qrow + 8), qA3 = ld2(qrow + 12);

        v8f s = {};
        s = WMMA_F32X4(qA0, kB0, s);
        s = WMMA_F32X4(qA1, kB1, s);
        s = WMMA_F32X4(qA2, kB2, s);
        s = WMMA_F32X4(qA3, kB3, s);

        float e[8];
        float tm = NEG_BIG;
#pragma unroll
        for (int r = 0; r < 8; ++r) {
            int jabs = j0 + r + rhi;
            float sv = s[r] * INV_SQRT_KS;
            e[r] = (jabs >= iabs) ? sv : NEG_BIG;
            tm = fmaxf(tm, e[r]);
        }
        float newm = fmaxf(m, tm);
        float zs = 0.0f;
#pragma unroll
        for (int r = 0; r < 8; ++r) zs += __expf(e[r] - newm);
        z = z * __expf(m - newm) + zs;   // garbage from all-masked phases is
        m = newm;                         // annihilated once a real max arrives
    }
    // merge the two half-columns (lanes l and l^16)
    float mo = __shfl_xor(m, 16, 32);
    float zo = __shfl_xor(z, 16, 32);
    float fm = fmaxf(m, mo);
    float fz = z * __expf(m - fm) + zo * __expf(mo - fm);
    if (lane < 16) {
        Mb[base + iabs] = fm;
        Zr[base + iabs] = 1.0f / fz;
    }
}

// ---------------------------------------------------------------------------
// Kernel 3: O[j,:] = sum_{i<=j} exp(s[j,i]/4 - M[i]) * (1/Z[i]) * V[i,:]
// One wave per (batch, j-tile). P tile goes D-layout -> LDS -> A-layout.
// ---------------------------------------------------------------------------
__global__ void attn_out_kernel(const float* __restrict__ qb, const float* __restrict__ kb,
                                const float* __restrict__ vb, const float* __restrict__ Mb,
                                const float* __restrict__ Zr, float* __restrict__ out) {
    __shared__ float pbuf[16 * 17];   // padded to dodge bank conflicts
    const int lane  = threadIdx.x;
    const int b     = blockIdx.x / NT_;
    const int jtile = blockIdx.x % NT_;
    const int j0    = jtile * 16;
    const int base  = b * T_;
    const int n     = lane & 15;
    const int koff  = (lane >> 4) * 2;
    const int rhi   = 8 * (lane >> 4);

    const float* qrow = qb + (size_t)(base + j0 + n) * HD_ + koff;
    v2f qA0 = ld2(qrow + 0), qA1 = ld2(qrow + 4), qA2 = ld2(qrow + 8), qA3 = ld2(qrow + 12);

    v8f oacc = {};
    for (int it = 0; it <= jtile; ++it) {
        const int i0 = it * 16;
        const float* krow = kb + (size_t)(base + i0 + n) * HD_ + koff;
        v2f kB0 = ld2(krow + 0), kB1 = ld2(krow + 4), kB2 = ld2(krow + 8), kB3 = ld2(krow + 12);

        v8f s = {};
        s = WMMA_F32X4(qA0, kB0, s);
        s = WMMA_F32X4(qA1, kB1, s);
        s = WMMA_F32X4(qA2, kB2, s);
        s = WMMA_F32X4(qA3, kB3, s);

        const float Mi  = Mb[base + i0 + n];
        const float rzi = Zr[base + i0 + n];

        __syncthreads();   // pbuf from previous iteration fully consumed
#pragma unroll
        for (int r = 0; r < 8; ++r) {
            int jabs = j0 + r + rhi;
            int iabs = i0 + n;
            float p = (jabs >= iabs) ? __expf(s[r] * INV_SQRT_KS - Mi) * rzi : 0.0f;
            pbuf[(r + rhi) * 17 + n] = p;   // pbuf[j][i]
        }
        __syncthreads();

        // A-layout chunks of P: element (m = n, k = kc + koff + r)
        const float* prow = pbuf + n * 17 + koff;
        v2f pA0 = { prow[0],  prow[1]  };
        v2f pA1 = { prow[4],  prow[5]  };
        v2f pA2 = { prow[8],  prow[9]  };
        v2f pA3 = { prow[12], prow[13] };

        // B-layout chunks of V: B[k=i][n=v] = vb[(base+i0+k)*16 + n]
        const float* vcol = vb + (size_t)(base + i0 + koff) * HD_ + n;
        v2f vB0 = { vcol[0 * HD_],  vcol[1 * HD_]  };
        v2f vB1 = { vcol[4 * HD_],  vcol[5 * HD_]  };
        v2f vB2 = { vcol[8 * HD_],  vcol[9 * HD_]  };
        v2f vB3 = { vcol[12 * HD_], vcol[13 * HD_] };

        oacc = WMMA_F32X4(pA0, vB0, oacc);
        oacc = WMMA_F32X4(pA1, vB1, oacc);
        oacc = WMMA_F32X4(pA2, vB2, oacc);
        oacc = WMMA_F32X4(pA3, vB3, oacc);
    }
#pragma unroll
    for (int r = 0; r < 8; ++r) {
        int row = base + j0 + r + rhi;
        out[(size_t)row * OUTC_ + C_ + n] = oacc[r];
    }
}

// ---------------------------------------------------------------------------
extern "C" void kernel_launch(void* const* d_in, const int* in_sizes, int n_in,
                              void* d_out, int out_size, void* d_ws, size_t ws_size,
                              hipStream_t stream) {
    (void)in_sizes; (void)n_in; (void)out_size; (void)ws_size;
    const float* x  = (const float*)d_in[0];
    const float* Wq = (const float*)d_in[1];
    const float* bq = (const float*)d_in[2];
    const float* Wk = (const float*)d_in[3];
    const float* bk = (const float*)d_in[4];
    const float* Wv = (const float*)d_in[5];
    const float* bv = (const float*)d_in[6];
    float* out = (float*)d_out;

    const int rows = B_ * T_;                     // 32768
    float* qb = (float*)d_ws;                     // rows*16
    float* kb = qb + (size_t)rows * HD_;          // rows*16
    float* vb = kb + (size_t)rows * HD_;          // rows*16
    float* Mb = vb + (size_t)rows * HD_;          // rows
    float* Zr = Mb + rows;                        // rows   (total ~6.4 MB)

    qkv_proj_copy_kernel<<<rows / 16, 32, 0, stream>>>(x, Wq, bq, Wk, bk, Wv, bv,
                                                       qb, kb, vb, out);
    colsoftmax_stats_kernel<<<B_ * NT_, 32, 0, stream>>>(qb, kb, Mb, Zr);
    attn_out_kernel<<<B_ * NT_, 32, 0, stream>>>(qb, kb, vb, Mb, Zr, out);
}